// AVT_VQVAE_Encoder_43044162240521
// MI455X (gfx1250) — compile-verified
//
#include <hip/hip_runtime.h>
#include <hip/hip_bf16.h>

typedef float v2f __attribute__((ext_vector_type(2)));
typedef float v8f __attribute__((ext_vector_type(8)));

#define N_ROWS   8192   // B*T
#define M_CODES  1024
#define DDIM     256
#define DTOT     768

// ---------------- row sum-of-squares ----------------
__global__ void rowsq_kernel(const float* __restrict__ src, float* __restrict__ dst,
                             int rowstride, int coloff) {
  __shared__ float red[256];
  const int row = blockIdx.x;
  const float v = src[(long long)row * rowstride + coloff + threadIdx.x];
  red[threadIdx.x] = v * v;
  __syncthreads();
  for (int s = 128; s > 0; s >>= 1) {
    if (threadIdx.x < s) red[threadIdx.x] += red[threadIdx.x + s];
    __syncthreads();
  }
  if (threadIdx.x == 0) dst[row] = red[0];
}

__device__ __forceinline__ void merge_softmax(float& m, float& s, float om, float os) {
  float nm = fmaxf(m, om);
  s = s * __expf(m - nm) + os * __expf(om - nm);
  m = nm;
}

// ---------------- fused distance GEMM + softmax + argmin ----------------
// grid (N_ROWS/16, 3 modalities), block 128 (4 waves). Each wave owns 16 of the
// 64 column tiles (interleaved), computed with V_WMMA_F32_16X16X4_F32.
__global__ __launch_bounds__(128)
void dist_kernel(const float* __restrict__ A0, const float* __restrict__ A1,
                 const float* __restrict__ A2, const float* __restrict__ E,
                 const float* __restrict__ xsq, const float* __restrict__ esq,
                 float* __restrict__ pH, int* __restrict__ idxout) {
  __shared__ _Float16 zbuf[16][M_CODES];          // 32 KB: z = -sqrt(dist)
  __shared__ float redM[4][16], redS[4][16], redD[4][16];
  __shared__ int   redI[4][16];
  __shared__ float finM[16], finSinv[16];

  const int mod  = blockIdx.y;
  const int n0   = blockIdx.x * 16;
  const int wave = threadIdx.x >> 5;
  const int lane = threadIdx.x & 31;
  const int lo   = lane & 15;
  const int hi   = lane >> 4;

  const float* X   = (mod == 0) ? A0 : ((mod == 1) ? A1 : A2);
  const int   coff = (mod == 0) ? DDIM : ((mod == 1) ? 0 : 2 * DDIM);

  const float* xrow = X + (long long)(n0 + lo) * DDIM;   // A-frag source row
  const float* xsqm = xsq + mod * N_ROWS + n0;
  const float* esqm = esq + mod * M_CODES;

  float xsqv[8];
#pragma unroll
  for (int j = 0; j < 8; ++j) xsqv[j] = xsqm[j + 8 * hi];  // C row = j + 8*hi

  float m8[8], s8[8], dmin8[8];
  int   imin8[8];
#pragma unroll
  for (int j = 0; j < 8; ++j) {
    m8[j] = -__builtin_inff(); s8[j] = 0.f;
    dmin8[j] = __builtin_inff(); imin8[j] = 0;
  }

  for (int ct = wave; ct < 64; ct += 4) {
    const int c = ct * 16 + lo;                       // this lane's output column
    const float* erow = E + (long long)c * DTOT + coff;
    v8f acc = {0.f, 0.f, 0.f, 0.f, 0.f, 0.f, 0.f, 0.f};
#pragma unroll 4
    for (int k = 0; k < DDIM; k += 4) {
      v2f a = *(const v2f*)(xrow + k + 2 * hi);       // A: row=lo, K = k+2*hi, k+2*hi+1
      v2f b = *(const v2f*)(erow + k + 2 * hi);       // B: col=lo, same K split
      acc = __builtin_amdgcn_wmma_f32_16x16x4_f32(false, a, false, b,
                                                  (short)0, acc, false, false);
    }
    const float esq_c = esqm[c];
#pragma unroll
    for (int j = 0; j < 8; ++j) {
      const int r = j + 8 * hi;
      float dist = fmaxf(xsqv[j] + esq_c - 2.0f * acc[j], 0.0f);
      float z = -sqrtf(dist);
      zbuf[r][c] = (_Float16)z;
      float nm = fmaxf(m8[j], z);
      s8[j] = s8[j] * __expf(m8[j] - nm) + __expf(z - nm);
      m8[j] = nm;
      if (dist < dmin8[j]) { dmin8[j] = dist; imin8[j] = c; }   // cols increase -> first-min kept
    }
  }

  // merge across the 16 lanes that share a row (xor masks stay within half-wave)
#pragma unroll
  for (int j = 0; j < 8; ++j) {
    float m = m8[j], s = s8[j], d = dmin8[j]; int ii = imin8[j];
    for (int off = 1; off < 16; off <<= 1) {
      float om = __shfl_xor(m, off, 32);
      float os = __shfl_xor(s, off, 32);
      float od = __shfl_xor(d, off, 32);
      int   oi = __shfl_xor(ii, off, 32);
      merge_softmax(m, s, om, os);
      if (od < d || (od == d && oi < ii)) { d = od; ii = oi; }
    }
    m8[j] = m; s8[j] = s; dmin8[j] = d; imin8[j] = ii;
  }
  if (lo == 0) {
#pragma unroll
    for (int j = 0; j < 8; ++j) {
      const int r = j + 8 * hi;
      redM[wave][r] = m8[j]; redS[wave][r] = s8[j];
      redD[wave][r] = dmin8[j]; redI[wave][r] = imin8[j];
    }
  }
  __syncthreads();

  if (threadIdx.x < 16) {
    const int r = threadIdx.x;
    float m = redM[0][r], s = redS[0][r], d = redD[0][r]; int ii = redI[0][r];
    for (int w = 1; w < 4; ++w) {
      merge_softmax(m, s, redM[w][r], redS[w][r]);
      float od = redD[w][r]; int oi = redI[w][r];
      if (od < d || (od == d && oi < ii)) { d = od; ii = oi; }
    }
    finM[r] = m; finSinv[r] = 1.0f / s;
    idxout[mod * N_ROWS + n0 + r] = ii;
  }
  __syncthreads();

  // pass 2: accumulate softmax probabilities into pH (sum over T; /T done later)
  const int b = n0 >> 8;                      // T=256, strip stays in one batch
  float* pHrow = pH + (long long)(mod * 32 + b) * M_CODES;
  for (int c = threadIdx.x; c < M_CODES; c += 128) {
    float sum = 0.f;
#pragma unroll
    for (int r = 0; r < 16; ++r)
      sum += __expf((float)zbuf[r][c] - finM[r]) * finSinv[r];
    atomicAdd(&pHrow[c], sum);
  }
}

// ---------------- gather quant / full outputs ----------------
__global__ void gather_kernel(const float* __restrict__ E, const int* __restrict__ idx,
                              float* __restrict__ out) {
  const int n = blockIdx.x, mod = blockIdx.y, t = threadIdx.x;
  const long long fullbase[3] = {6291456LL, 0LL, 12582912LL};        // a, v, t slots
  const long long qbase[3]    = {20971520LL, 18874368LL, 23068672LL};
  const int coff[3]           = {DDIM, 0, 2 * DDIM};
  const int id = idx[mod * N_ROWS + n];
  const float4* src = (const float4*)(E + (long long)id * DTOT);
  float4* dst = (float4*)(out + fullbase[mod] + (long long)n * DTOT);
  dst[t] = src[t];                                                    // 192 * 4 = 768
  if (t < 64) {
    const float4* sq = (const float4*)(E + (long long)id * DTOT + coff[mod]);
    float4* dq = (float4*)(out + qbase[mod] + (long long)n * DDIM);
    dq[t] = sq[t];                                                    // 64 * 4 = 256
  }
}

// ---------------- Lcmcm losses ----------------
__global__ void lcmcm_kernel(const float* __restrict__ pH, float* __restrict__ losses) {
  __shared__ float red[1024];
  __shared__ float li[32];
  const int p = blockIdx.x;                    // 0: av, 1: at, 2: tv
  const int ia = (p == 2) ? 2 : 0;
  const int ib = (p == 1) ? 2 : 1;
  const float* pA = pH + (long long)ia * 32 * M_CODES;
  const float* pB = pH + (long long)ib * 32 * M_CODES;
  const int i = threadIdx.x >> 5, j = threadIdx.x & 31;
  const float inv = 1.0f / 256.0f;
  float s = 0.f;
  for (int m = 0; m < M_CODES; ++m) {
    float ai = pA[i * M_CODES + m] * inv, bj = pB[j * M_CODES + m] * inv;
    float bi = pB[i * M_CODES + m] * inv, aj = pA[j * M_CODES + m] * inv;
    s += ai * __logf(bj + 1e-10f) + bi * __logf(aj + 1e-10f);
  }
  red[threadIdx.x] = s;
  __syncthreads();
  for (int st = 512; st > 0; st >>= 1) {       // global min of S
    if (threadIdx.x < st) red[threadIdx.x] = fminf(red[threadIdx.x], red[threadIdx.x + st]);
    __syncthreads();
  }
  const float smin = red[0];
  const float e = __expf(s - smin);
  float rowsum = e;
  for (int off = 16; off > 0; off >>= 1) rowsum += __shfl_xor(rowsum, off, 32);
  if (j == i) li[i] = __logf(e / (rowsum + 1e-5f));
  __syncthreads();
  if (threadIdx.x == 0) {
    float acc = 0.f;
    for (int r = 0; r < 32; ++r) acc += li[r];
    losses[p] = -acc / 32.0f;
  }
}

// ---------------- per-batch mode over T ----------------
__global__ void mode_kernel(const int* __restrict__ idx, int* __restrict__ modes) {
  __shared__ int counts[M_CODES];
  __shared__ int redk[256];
  const int b = blockIdx.x, mod = blockIdx.y, t = threadIdx.x;
  for (int m = t; m < M_CODES; m += 256) counts[m] = 0;
  __syncthreads();
  atomicAdd(&counts[idx[mod * N_ROWS + b * 256 + t]], 1);
  __syncthreads();
  int best = -1;
  for (int m = t; m < M_CODES; m += 256)
    best = max(best, (counts[m] << 10) | (1023 - m));   // argmax, first occurrence
  redk[t] = best;
  __syncthreads();
  for (int st = 128; st > 0; st >>= 1) {
    if (t < st) redk[t] = max(redk[t], redk[t + st]);
    __syncthreads();
  }
  if (t == 0) modes[mod * 32 + b] = 1023 - (redk[0] & 1023);
}

__global__ void equal_kernel(const int* __restrict__ modes, float* __restrict__ outv) {
  const int b = threadIdx.x;
  const int ma = modes[b], mv = modes[32 + b], mt = modes[64 + b];
  int e = (ma == mv && ma == mt) ? 1 : 0;
  for (int off = 16; off > 0; off >>= 1) e += __shfl_xor(e, off, 32);
  if (b == 0) outv[0] = (float)e;
}

extern "C" void kernel_launch(void* const* d_in, const int* in_sizes, int n_in,
                              void* d_out, int out_size, void* d_ws, size_t ws_size,
                              hipStream_t stream) {
  (void)in_sizes; (void)n_in; (void)out_size; (void)ws_size;
  const float* audio = (const float*)d_in[0];
  const float* video = (const float*)d_in[1];
  const float* text  = (const float*)d_in[2];
  const float* E     = (const float*)d_in[3];
  float* out = (float*)d_out;

  char* ws = (char*)d_ws;
  int*   idx   = (int*)ws;                                  //  3*8192 int
  float* pH    = (float*)(ws + 98304);                      //  3*32*1024 f32
  float* xsq   = (float*)(ws + 98304 + 393216);             //  3*8192 f32
  float* esq   = (float*)(ws + 98304 + 393216 + 98304);     //  3*1024 f32
  int*   modes = (int*)(ws + 98304 + 393216 + 98304 + 12288);

  hipMemsetAsync(pH, 0, 3 * 32 * M_CODES * sizeof(float), stream);

  rowsq_kernel<<<N_ROWS, 256, 0, stream>>>(audio, xsq,             DDIM, 0);
  rowsq_kernel<<<N_ROWS, 256, 0, stream>>>(video, xsq + N_ROWS,    DDIM, 0);
  rowsq_kernel<<<N_ROWS, 256, 0, stream>>>(text,  xsq + 2*N_ROWS,  DDIM, 0);
  rowsq_kernel<<<M_CODES, 256, 0, stream>>>(E, esq,               DTOT, DDIM);     // a slice
  rowsq_kernel<<<M_CODES, 256, 0, stream>>>(E, esq + M_CODES,     DTOT, 0);        // v slice
  rowsq_kernel<<<M_CODES, 256, 0, stream>>>(E, esq + 2*M_CODES,   DTOT, 2*DDIM);   // t slice

  dist_kernel<<<dim3(N_ROWS / 16, 3), 128, 0, stream>>>(audio, video, text, E,
                                                        xsq, esq, pH, idx);
  gather_kernel<<<dim3(N_ROWS, 3), 192, 0, stream>>>(E, idx, out);
  lcmcm_kernel<<<3, 1024, 0, stream>>>(pH, out + 25165824);
  mode_kernel<<<dim3(32, 3), 256, 0, stream>>>(idx, modes);
  equal_kernel<<<1, 32, 0, stream>>>(modes, out + 25165827);
}